// SegSmall_Features_Generator_3642132267203
// MI455X (gfx1250) — compile-verified
//
#include <hip/hip_runtime.h>
#include <hip/hip_bf16.h>
#include <math.h>

// CDNA5 / gfx1250 implementation of the SegSmall features generator.
// f32 WMMA (V_WMMA_F32_16X16X4_F32) is used for the per-neighbor MLP GEMMs.

typedef float v2f __attribute__((ext_vector_type(2)));
typedef float v8f __attribute__((ext_vector_type(8)));

#define INF_F 3.402823466e+38f
#define MLP_WAVES 4

// --------------------------------------------------------------------------
// KNN (exact, jax.lax.top_k tie-break: ascending d2, then ascending index)
// One wave32 per query. Per-lane sorted top-K in registers, wave-merge via
// u64-key shuffle min-reduction. Also emits normalized rel vectors.
// --------------------------------------------------------------------------
template <int K>
__global__ void knn_rel_kernel(const float* __restrict__ pts, int Nsrc, int npts,
                               int Q, int* __restrict__ idxOut,
                               float* __restrict__ relOut) {
  const int lane = threadIdx.x & 31;
  const int q = blockIdx.x * (blockDim.x >> 5) + (threadIdx.x >> 5);
  if (q >= Q) return;  // wave-uniform
  const int b = q / npts;
  const int p = q - b * npts;
  const float* P = pts + (size_t)b * 8192 * 3;  // input_pts batch stride
  const float qx = P[p * 3 + 0], qy = P[p * 3 + 1], qz = P[p * 3 + 2];

  float bd[K];
  int bi[K];
#pragma unroll
  for (int t = 0; t < K; ++t) { bd[t] = INF_F; bi[t] = 0x7FFFFFFF; }

  for (int j = lane; j < Nsrc; j += 32) {
    float dx = P[j * 3 + 0] - qx, dy = P[j * 3 + 1] - qy, dz = P[j * 3 + 2] - qz;
    float d2 = dx * dx + dy * dy + dz * dz;
    if (d2 < bd[K - 1] || (d2 == bd[K - 1] && j < bi[K - 1])) {
      float cd = d2; int ci = j;
#pragma unroll
      for (int t = 0; t < K; ++t) {
        bool sw = (cd < bd[t]) || (cd == bd[t] && ci < bi[t]);
        float td = sw ? bd[t] : cd; int ti = sw ? bi[t] : ci;
        bd[t] = sw ? cd : bd[t]; bi[t] = sw ? ci : bi[t];
        cd = td; ci = ti;
      }
    }
  }

  // K rounds of wave-wide min-selection over sorted per-lane heads.
  int myNb = 0;
#pragma unroll
  for (int s = 0; s < K; ++s) {
    unsigned long long key =
        ((unsigned long long)__float_as_uint(bd[0]) << 32) | (unsigned int)bi[0];
    unsigned long long k = key;
#pragma unroll
    for (int off = 16; off > 0; off >>= 1) {
      unsigned long long o = __shfl_xor(k, off, 32);
      k = (o < k) ? o : k;
    }
    int selIdx = (int)(unsigned int)(k & 0xFFFFFFFFull);
    if (lane == s) myNb = selIdx;
    if (lane == 0) idxOut[q * K + s] = selIdx;
    if (key == k) {  // unique winner pops its head
#pragma unroll
      for (int t = 0; t < K - 1; ++t) { bd[t] = bd[t + 1]; bi[t] = bi[t + 1]; }
      bd[K - 1] = INF_F; bi[K - 1] = 0x7FFFFFFF;
    }
  }

  // rel = nb_pts - query, normalized by sqrt(max ||rel||^2) (0 -> 1)
  float rx = 0.f, ry = 0.f, rz = 0.f, dd = 0.f;
  if (lane < K) {
    rx = P[myNb * 3 + 0] - qx;
    ry = P[myNb * 3 + 1] - qy;
    rz = P[myNb * 3 + 2] - qz;
    dd = rx * rx + ry * ry + rz * rz;
  }
#pragma unroll
  for (int off = 16; off > 0; off >>= 1) dd = fmaxf(dd, __shfl_xor(dd, off, 32));
  float maxi = sqrtf(dd);
  if (maxi == 0.0f) maxi = 1.0f;
  float inv = 1.0f / maxi;
  if (lane < K) {
    relOut[(size_t)(q * K + lane) * 3 + 0] = rx * inv;
    relOut[(size_t)(q * K + lane) * 3 + 1] = ry * inv;
    relOut[(size_t)(q * K + lane) * 3 + 2] = rz * inv;
  }
}

// --------------------------------------------------------------------------
// MLP over R = B*npts*K rows: d(48) ->32 relu ->16 relu ->16 relu.
// One wave32 per 16-row tile; f32 WMMA 16x16x4; inter-layer transpose via LDS
// (DS ops are in-order within a wave; s_wait_dscnt fences the load data).
// d[row][j] = rel[row][j>>4] - centers[j]   (j = dim*16 + center)
// --------------------------------------------------------------------------
__global__ void mlp_wmma_kernel(const float* __restrict__ rel, int R,
                                const float* __restrict__ centers,
                                const float* __restrict__ w1, const float* __restrict__ b1,
                                const float* __restrict__ w2, const float* __restrict__ b2,
                                const float* __restrict__ w3, const float* __restrict__ b3,
                                float* __restrict__ h3out) {
  __shared__ float sC[48];
  __shared__ float sW1[48 * 32], sB1[32];
  __shared__ float sW2[32 * 16], sB2[16];
  __shared__ float sW3[16 * 16], sB3[16];
  __shared__ float sH1[MLP_WAVES][16 * 32];
  __shared__ float sH2[MLP_WAVES][16 * 16];

  for (int t = threadIdx.x; t < 48; t += blockDim.x) sC[t] = centers[t];
  for (int t = threadIdx.x; t < 48 * 32; t += blockDim.x) sW1[t] = w1[t];
  for (int t = threadIdx.x; t < 32; t += blockDim.x) sB1[t] = b1[t];
  for (int t = threadIdx.x; t < 32 * 16; t += blockDim.x) sW2[t] = w2[t];
  for (int t = threadIdx.x; t < 16; t += blockDim.x) sB2[t] = b2[t];
  for (int t = threadIdx.x; t < 16 * 16; t += blockDim.x) sW3[t] = w3[t];
  for (int t = threadIdx.x; t < 16; t += blockDim.x) sB3[t] = b3[t];
  __syncthreads();

  const int lane = threadIdx.x & 31;
  const int wv = threadIdx.x >> 5;
  const int tile = blockIdx.x * MLP_WAVES + wv;
  const int row0 = tile * 16;
  if (row0 >= R) return;  // wave-uniform; EXEC stays all-ones for WMMA

  const int m = lane & 15;   // M for A / N-column for B,C,D
  const int hi = lane >> 4;  // half-wave select (K+2 for A/B, M+8 for C/D)

  const float relv0 = rel[(size_t)(row0 + m) * 3 + 0];
  const float relv1 = rel[(size_t)(row0 + m) * 3 + 1];
  const float relv2 = rel[(size_t)(row0 + m) * 3 + 2];

  // ---- layer 1: (16x48) @ (48x32) + b1, relu; two N-tiles of 16 ----
  v8f acc0, acc1;
#pragma unroll
  for (int r = 0; r < 8; ++r) { acc0[r] = sB1[m]; acc1[r] = sB1[16 + m]; }
#pragma unroll
  for (int kc = 0; kc < 12; ++kc) {
    const int kb = kc * 4;
    const int j0 = kb + 2 * hi;  // K-chunk never crosses a 16-boundary
    const float rv = (kb < 16) ? relv0 : ((kb < 32) ? relv1 : relv2);
    v2f a; a.x = rv - sC[j0]; a.y = rv - sC[j0 + 1];
    v2f bb0; bb0.x = sW1[j0 * 32 + m]; bb0.y = sW1[(j0 + 1) * 32 + m];
    acc0 = __builtin_amdgcn_wmma_f32_16x16x4_f32(false, a, false, bb0, (short)0,
                                                 acc0, false, false);
    v2f bb1; bb1.x = sW1[j0 * 32 + 16 + m]; bb1.y = sW1[(j0 + 1) * 32 + 16 + m];
    acc1 = __builtin_amdgcn_wmma_f32_16x16x4_f32(false, a, false, bb1, (short)0,
                                                 acc1, false, false);
  }
#pragma unroll
  for (int r = 0; r < 8; ++r) { acc0[r] = fmaxf(acc0[r], 0.f); acc1[r] = fmaxf(acc1[r], 0.f); }
  float* H1 = sH1[wv];
#pragma unroll
  for (int r = 0; r < 8; ++r) {
    H1[(r + 8 * hi) * 32 + m] = acc0[r];
    H1[(r + 8 * hi) * 32 + 16 + m] = acc1[r];
  }
  asm volatile("s_wait_dscnt 0" ::: "memory");

  // ---- layer 2: (16x32) @ (32x16) + b2, relu ----
  v8f acc2;
#pragma unroll
  for (int r = 0; r < 8; ++r) acc2[r] = sB2[m];
#pragma unroll
  for (int kc = 0; kc < 8; ++kc) {
    const int j0 = kc * 4 + 2 * hi;
    v2f a; a.x = H1[m * 32 + j0]; a.y = H1[m * 32 + j0 + 1];
    v2f bb; bb.x = sW2[j0 * 16 + m]; bb.y = sW2[(j0 + 1) * 16 + m];
    acc2 = __builtin_amdgcn_wmma_f32_16x16x4_f32(false, a, false, bb, (short)0,
                                                 acc2, false, false);
  }
#pragma unroll
  for (int r = 0; r < 8; ++r) acc2[r] = fmaxf(acc2[r], 0.f);
  float* H2 = sH2[wv];
#pragma unroll
  for (int r = 0; r < 8; ++r) H2[(r + 8 * hi) * 16 + m] = acc2[r];
  asm volatile("s_wait_dscnt 0" ::: "memory");

  // ---- layer 3: (16x16) @ (16x16) + b3, relu ----
  v8f acc3;
#pragma unroll
  for (int r = 0; r < 8; ++r) acc3[r] = sB3[m];
#pragma unroll
  for (int kc = 0; kc < 4; ++kc) {
    const int j0 = kc * 4 + 2 * hi;
    v2f a; a.x = H2[m * 16 + j0]; a.y = H2[m * 16 + j0 + 1];
    v2f bb; bb.x = sW3[j0 * 16 + m]; bb.y = sW3[(j0 + 1) * 16 + m];
    acc3 = __builtin_amdgcn_wmma_f32_16x16x4_f32(false, a, false, bb, (short)0,
                                                 acc3, false, false);
  }
#pragma unroll
  for (int r = 0; r < 8; ++r) {
    h3out[(size_t)(row0 + r + 8 * hi) * 16 + m] = fmaxf(acc3[r], 0.f);
  }
}

// --------------------------------------------------------------------------
// f[c][m] = sum_k feats[idx[k]][c] * h3[k][m];  out[o] = (f . W[:,:,o]) / K
// One block per point; f staged in LDS.
// --------------------------------------------------------------------------
__global__ void einsum_proj_kernel(const float* __restrict__ feats, int Nsrc,
                                   int cin, int cout, int K,
                                   const int* __restrict__ idx,
                                   const float* __restrict__ h3,
                                   const float* __restrict__ W, int npts,
                                   float* __restrict__ outraw) {
  __shared__ float fl[96 * 16];
  __shared__ float sH[16 * 16];
  __shared__ int sIdx[16];
  const int q = blockIdx.x;
  const int b = q / npts;
  const int tid = threadIdx.x;
  if (tid < K) sIdx[tid] = idx[q * K + tid];
  for (int t = tid; t < K * 16; t += blockDim.x)
    sH[t] = h3[(size_t)q * K * 16 + t];
  __syncthreads();

  const int F = cin * 16;
  for (int t = tid; t < F; t += blockDim.x) {
    const int c = t >> 4, mm = t & 15;
    float s = 0.f;
    for (int k = 0; k < K; ++k)
      s += feats[((size_t)b * Nsrc + sIdx[k]) * cin + c] * sH[k * 16 + mm];
    fl[t] = s;
  }
  __syncthreads();

  const float invK = 1.0f / (float)K;
  for (int o = tid; o < cout; o += blockDim.x) {
    float acc = 0.f;
    for (int t = 0; t < F; ++t) acc += fl[t] * W[(size_t)t * cout + o];
    outraw[(size_t)q * cout + o] = acc * invK;
  }
}

// --------------------------------------------------------------------------
// BatchNorm over (B, npts) then ReLU.
// --------------------------------------------------------------------------
__global__ void bn_stats_kernel(const float* __restrict__ x, int rows, int C,
                                float* __restrict__ stats) {
  __shared__ float ssum[256], ssq[256];
  const int ch = blockIdx.x;
  float s = 0.f, s2 = 0.f;
  for (int r = threadIdx.x; r < rows; r += blockDim.x) {
    const float v = x[(size_t)r * C + ch];
    s += v; s2 += v * v;
  }
  ssum[threadIdx.x] = s; ssq[threadIdx.x] = s2;
  __syncthreads();
  for (int off = blockDim.x >> 1; off > 0; off >>= 1) {
    if ((int)threadIdx.x < off) {
      ssum[threadIdx.x] += ssum[threadIdx.x + off];
      ssq[threadIdx.x] += ssq[threadIdx.x + off];
    }
    __syncthreads();
  }
  if (threadIdx.x == 0) {
    const float mean = ssum[0] / (float)rows;
    stats[ch] = mean;
    stats[C + ch] = ssq[0] / (float)rows - mean * mean;
  }
}

__global__ void bn_apply_kernel(const float* __restrict__ x, int n, int C,
                                const float* __restrict__ stats,
                                const float* __restrict__ gamma,
                                const float* __restrict__ beta,
                                float* __restrict__ out) {
  const int t = blockIdx.x * blockDim.x + threadIdx.x;
  if (t >= n) return;
  const int ch = t % C;
  const float mean = stats[ch], var = stats[C + ch];
  const float v =
      (x[t] - mean) * (1.0f / sqrtf(var + 1e-5f)) * gamma[ch] + beta[ch];
  out[t] = fmaxf(v, 0.f);
}

__global__ void pts_copy_kernel(const float* __restrict__ pts, int npts, int n,
                                float* __restrict__ out) {
  const int t = blockIdx.x * blockDim.x + threadIdx.x;
  if (t >= n) return;
  const int per = npts * 3;
  const int b = t / per;
  out[t] = pts[(size_t)b * 8192 * 3 + (t - b * per)];
}

// --------------------------------------------------------------------------
// Host orchestration. Input order = recursive insertion-order flattening of
// setup_inputs(): x, input_pts, cv2..cv6 (weight, centers, l1_w, l1_b, l2_w,
// l2_b, l3_w, l3_b), bn2..bn6 (gamma, beta).
// --------------------------------------------------------------------------
extern "C" void kernel_launch(void* const* d_in, const int* in_sizes, int n_in,
                              void* d_out, int out_size, void* d_ws,
                              size_t ws_size, hipStream_t stream) {
  (void)in_sizes; (void)n_in; (void)out_size; (void)ws_size;
  const float* x = (const float*)d_in[0];
  const float* pts = (const float*)d_in[1];
  static const int CV[5] = {2, 10, 18, 26, 34};
  static const int BNI[5] = {42, 44, 46, 48, 50};

  float* out = (float*)d_out;
  float* x6 = out + 0;      float* p6 = out + 6144;
  float* x5 = out + 6336;   float* p5 = out + 18624;
  float* x4 = out + 19008;  float* p4 = out + 68160;
  float* x3 = out + 69696;  float* p3 = out + 168000;
  float* x2 = out + 174144; float* p2 = out + 567360;

  char* ws = (char*)d_ws;
  int* idxB = (int*)(ws + 0);                 // <= 512 KB
  float* relB = (float*)(ws + 524288);        // <= 1.5 MB
  float* h3B = (float*)(ws + 2097152);        // <= 8 MB
  float* rawB = (float*)(ws + 10485760);      // <= 1.5 MB
  float* statB = (float*)(ws + 12058624);     // <= 768 B

  struct Cfg {
    int Nsrc, npts, K, cin, cout;
    const float* feats; float* xout; float* pout;
  };
  const Cfg cfg[5] = {
      {8192, 1024, 16, 3, 48, x, x2, p2},
      {1024, 256, 16, 48, 48, x2, x3, p3},
      {256, 64, 8, 48, 96, x3, x4, p4},
      {64, 16, 8, 96, 96, x4, x5, p5},
      {16, 8, 4, 96, 96, x5, x6, p6},
  };

  for (int L = 0; L < 5; ++L) {
    const Cfg& c = cfg[L];
    const float* W = (const float*)d_in[CV[L] + 0];
    const float* cen = (const float*)d_in[CV[L] + 1];
    const float* w1 = (const float*)d_in[CV[L] + 2];
    const float* b1 = (const float*)d_in[CV[L] + 3];
    const float* w2 = (const float*)d_in[CV[L] + 4];
    const float* b2 = (const float*)d_in[CV[L] + 5];
    const float* w3 = (const float*)d_in[CV[L] + 6];
    const float* b3 = (const float*)d_in[CV[L] + 7];
    const float* gm = (const float*)d_in[BNI[L] + 0];
    const float* bt = (const float*)d_in[BNI[L] + 1];

    const int Q = 8 * c.npts;
    const int R = Q * c.K;

    const int knnBlocks = (Q + 3) / 4;  // 4 waves/block
    if (c.K == 16)
      knn_rel_kernel<16><<<knnBlocks, 128, 0, stream>>>(pts, c.Nsrc, c.npts, Q, idxB, relB);
    else if (c.K == 8)
      knn_rel_kernel<8><<<knnBlocks, 128, 0, stream>>>(pts, c.Nsrc, c.npts, Q, idxB, relB);
    else
      knn_rel_kernel<4><<<knnBlocks, 128, 0, stream>>>(pts, c.Nsrc, c.npts, Q, idxB, relB);

    const int tiles = R / 16;  // R is always a multiple of 16
    mlp_wmma_kernel<<<(tiles + MLP_WAVES - 1) / MLP_WAVES, MLP_WAVES * 32, 0,
                      stream>>>(relB, R, cen, w1, b1, w2, b2, w3, b3, h3B);

    einsum_proj_kernel<<<Q, 128, 0, stream>>>(c.feats, c.Nsrc, c.cin, c.cout,
                                              c.K, idxB, h3B, W, c.npts, rawB);

    bn_stats_kernel<<<c.cout, 256, 0, stream>>>(rawB, Q, c.cout, statB);
    const int n = Q * c.cout;
    bn_apply_kernel<<<(n + 255) / 256, 256, 0, stream>>>(rawB, n, c.cout, statB,
                                                         gm, bt, c.xout);

    const int np3 = 8 * c.npts * 3;
    pts_copy_kernel<<<(np3 + 255) / 256, 256, 0, stream>>>(pts, c.npts, np3, c.pout);
  }
}